// PromptResampler_16406775070867
// MI455X (gfx1250) — compile-verified
//
#include <hip/hip_runtime.h>
#include <hip/hip_bf16.h>

#define B_      2
#define T_      1024
#define D_      1024
#define H_      16
#define KV_     4
#define HD_     64
#define CACHE_  1024
#define S_      2048
#define MAXSEQ_ 4096

typedef float v2f __attribute__((ext_vector_type(2)));
typedef float v8f __attribute__((ext_vector_type(8)));

__device__ __forceinline__ v8f wmma4(v2f a, v2f b, v8f c) {
  // D = A(16x4 f32) * B(4x16 f32) + C(16x16 f32)
  return __builtin_amdgcn_wmma_f32_16x16x4_f32(
      /*neg_a=*/false, a, /*neg_b=*/false, b,
      /*c_mod=*/(short)0, c, /*reuse_a=*/false, /*reuse_b=*/false);
}

// Async global->LDS B128 (CDNA5 TDM-class path, tracked by ASYNCcnt).
// LDS aperture: flat addr[31:0] == LDS byte offset.
__device__ __forceinline__ unsigned lds_off(const void* p) {
  return (unsigned)(unsigned long long)p;
}
__device__ __forceinline__ void async_ld_b128(unsigned lds, const void* gaddr) {
  asm volatile("global_load_async_to_lds_b128 %0, %1, off"
               :: "v"(lds), "v"(gaddr) : "memory");
}
__device__ __forceinline__ void wait_async_le(int n) {
  if (n == 0)      asm volatile("s_wait_asynccnt 0x0" ::: "memory");
  else if (n == 2) asm volatile("s_wait_asynccnt 0x2" ::: "memory");
  else             asm volatile("s_wait_asynccnt 0x4" ::: "memory");
}

// ---------------------------------------------------------------------------
// Generic fp32 WMMA GEMM: C[M,N] = A[M,K] * B[K,N], row-major.
// Block = 256 threads (8 waves). Block tile 64x64, wave tile 16x32.
// Double-buffered LDS staged via async global->LDS; prefetch next k-slab.
// Requires M%64==0, N%64==0, K%16==0 (true for all our calls).
// ---------------------------------------------------------------------------
__global__ __launch_bounds__(256) void wmma_gemm_f32(
    const float* __restrict__ A, const float* __restrict__ Bm,
    float* __restrict__ C, int M, int N, int K)
{
  __shared__ float As[2][64][20];   // [m][k], padded (stride 80B, 16B aligned)
  __shared__ float Bs[2][16][68];   // [k][n], padded (stride 272B, 16B aligned)

  const int tid   = threadIdx.x;
  const int wave  = tid >> 5;
  const int lane  = tid & 31;
  const int waveM = wave & 3;          // 4 row-groups of 16
  const int waveN = wave >> 2;         // 2 col-groups of 32
  const int n16   = lane & 15;
  const int half  = lane >> 4;
  const int kb    = half * 2;

  const int rowBase = blockIdx.y * 64;
  const int colBase = blockIdx.x * 64;

  // cooperative-load indices
  const int lAr = tid >> 2;            // 0..63
  const int lAc = (tid & 3) * 4;       // 0,4,8,12
  const int lBr = tid >> 4;            // 0..15
  const int lBc = (tid & 15) * 4;      // 0..60

  auto issue = [&](int k0, int buf) {
    async_ld_b128(lds_off(&As[buf][lAr][lAc]),
                  A + (size_t)(rowBase + lAr) * K + k0 + lAc);
    async_ld_b128(lds_off(&Bs[buf][lBr][lBc]),
                  Bm + (size_t)(k0 + lBr) * N + colBase + lBc);
  };

  v8f acc0 = {}, acc1 = {};

  issue(0, 0);
  int it = 0;
  for (int k0 = 0; k0 < K; k0 += 16, ++it) {
    const int buf = it & 1;
    if (k0 + 16 < K) { issue(k0 + 16, buf ^ 1); wait_async_le(2); }
    else             { wait_async_le(0); }
    __syncthreads();
#pragma unroll
    for (int kk = 0; kk < 4; kk++) {
      const int kd = kk * 4 + kb;
      v2f a;  a.x  = As[buf][waveM * 16 + n16][kd];
              a.y  = As[buf][waveM * 16 + n16][kd + 1];
      v2f b0; b0.x = Bs[buf][kd][waveN * 32 + n16];
              b0.y = Bs[buf][kd + 1][waveN * 32 + n16];
      v2f b1; b1.x = Bs[buf][kd][waveN * 32 + 16 + n16];
              b1.y = Bs[buf][kd + 1][waveN * 32 + 16 + n16];
      acc0 = wmma4(a, b0, acc0);
      acc1 = wmma4(a, b1, acc1);
    }
    __syncthreads();
  }

#pragma unroll
  for (int r = 0; r < 8; r++) {
    const int m  = rowBase + waveM * 16 + r + 8 * half;
    const int n0 = colBase + waveN * 32 + n16;
    C[(size_t)m * N + n0]      = acc0[r];
    C[(size_t)m * N + n0 + 16] = acc1[r];
  }
}

// ---------------------------------------------------------------------------
// RoPE helpers (double-precision angle -> fp32 reference-grade accuracy)
// ---------------------------------------------------------------------------
__device__ __forceinline__ void rope_cs(int p, int i, float& c, float& s) {
  const double inv_freq = exp(-(double)(2 * i) * 9.210340371976184 / 64.0); // ln(10000)
  double sd, cd;
  sincos((double)p * inv_freq, &sd, &cd);
  c = (float)cd; s = (float)sd;
}

// Q RoPE in place.  idx over B*T*H*32 = 1,048,576
__global__ __launch_bounds__(256) void rope_q_kernel(
    float* __restrict__ Q, const int* __restrict__ pos)
{
  const int idx = blockIdx.x * blockDim.x + threadIdx.x;
  const int i  = idx & 31;
  const int hh = (idx >> 5) & 15;
  const int t  = (idx >> 9) & (T_ - 1);
  const int b  = idx >> 19;
  int p = pos[b * T_ + t] + t + CACHE_;
  p = min(max(p, 0), MAXSEQ_ - 1);
  float c, s; rope_cs(p, i, c, s);
  const size_t base = (size_t)(b * T_ + t) * D_ + hh * HD_;
  const float q1 = Q[base + i];
  const float q2 = Q[base + i + 32];
  Q[base + i]      = q1 * c - q2 * s;
  Q[base + i + 32] = q2 * c + q1 * s;
}

// K RoPE + scatter, V scatter into output K/V at rows [CACHE_, S_).
// idx over B*T*KV*32 = 262,144
__global__ __launch_bounds__(256) void rope_kv_kernel(
    const float* __restrict__ Kn, const float* __restrict__ Vn,
    const int* __restrict__ pos,
    float* __restrict__ ok, float* __restrict__ ov)
{
  const int idx = blockIdx.x * blockDim.x + threadIdx.x;
  const int i  = idx & 31;
  const int kv = (idx >> 5) & 3;
  const int t  = (idx >> 7) & (T_ - 1);
  const int b  = idx >> 17;
  int p = pos[b * T_ + t] + t + CACHE_;
  p = min(max(p, 0), MAXSEQ_ - 1);
  float c, s; rope_cs(p, i, c, s);
  const size_t inrow = (size_t)(b * T_ + t) * (KV_ * HD_) + kv * HD_;
  const float k1 = Kn[inrow + i];
  const float k2 = Kn[inrow + i + 32];
  const size_t orow = ((size_t)(b * KV_ + kv) * S_ + CACHE_ + t) * HD_;
  ok[orow + i]      = k1 * c - k2 * s;
  ok[orow + i + 32] = k2 * c + k1 * s;
  ov[orow + i]      = Vn[inrow + i];
  ov[orow + i + 32] = Vn[inrow + i + 32];
}

// Copy caches into output K/V rows [0, CACHE_).  idx over B*KV*CACHE*HD = 524,288
__global__ __launch_bounds__(256) void cache_copy_kernel(
    const float* __restrict__ kc, const float* __restrict__ vc,
    float* __restrict__ ok, float* __restrict__ ov)
{
  const int idx = blockIdx.x * blockDim.x + threadIdx.x;
  const int bk  = idx >> 16;                 // CACHE_*HD_ = 65536
  const int rem = idx & 65535;
  ok[(size_t)bk * (S_ * HD_) + rem] = kc[idx];
  ov[(size_t)bk * (S_ * HD_) + rem] = vc[idx];
}

// ---------------------------------------------------------------------------
// Flash attention, fp32 WMMA. grid(8, H_, B_), 256 threads = 8 waves.
// Each block: 128 q-rows (16/wave) for one (b, h); iterates S in blocks of 32.
// K/V tiles double-buffered in LDS via async global->LDS with prefetch.
// Writes attention output back into QO (aliasing Q; regions disjoint/RAW-safe).
// ---------------------------------------------------------------------------
__global__ __launch_bounds__(256) void attn_fwd_kernel(
    float* __restrict__ QO, const float* __restrict__ Kt, const float* __restrict__ Vt)
{
  __shared__ float Ks[2][32][68];
  __shared__ float Vs[2][32][68];
  __shared__ float Ps[8][16][36];    // per-wave P tile (C-layout -> A-layout relay)

  const int tileQ = blockIdx.x;      // 0..7
  const int h     = blockIdx.y;      // 0..15
  const int b     = blockIdx.z;      // 0..1
  const int kvh   = h >> 2;          // n_rep = 4, contiguous grouping
  const int tid   = threadIdx.x;
  const int wave  = tid >> 5;
  const int lane  = tid & 31;
  const int n16   = lane & 15;
  const int half  = lane >> 4;
  const int kb    = half * 2;
  const int t0    = tileQ * 128;

  const float* kbase = Kt + (size_t)(b * KV_ + kvh) * S_ * HD_;
  const float* vbase = Vt + (size_t)(b * KV_ + kvh) * S_ * HD_;

  // Q as pre-shaped A fragments: 16 fragments cover hd=64 (K=4 each)
  float2 qa[16];
  {
    const float* qrow = QO + (size_t)(b * T_ + t0 + wave * 16 + n16) * D_ + h * HD_ + kb;
#pragma unroll
    for (int kk = 0; kk < 16; kk++) qa[kk] = *(const float2*)(qrow + kk * 4);
  }

  v8f acc[4];
  {
    v8f z = {};
#pragma unroll
    for (int ct = 0; ct < 4; ct++) acc[ct] = z;
  }
  float m_i[8], l_i[8];
#pragma unroll
  for (int r = 0; r < 8; r++) { m_i[r] = -1e30f; l_i[r] = 0.0f; }

  const int lr0 = tid >> 4;          // 0..15
  const int c0  = (tid & 15) * 4;    // 0..60

  auto issue_tile = [&](int s0, int buf) {
    async_ld_b128(lds_off(&Ks[buf][lr0][c0]),      kbase + (size_t)(s0 + lr0)      * HD_ + c0);
    async_ld_b128(lds_off(&Ks[buf][16 + lr0][c0]), kbase + (size_t)(s0 + 16 + lr0) * HD_ + c0);
    async_ld_b128(lds_off(&Vs[buf][lr0][c0]),      vbase + (size_t)(s0 + lr0)      * HD_ + c0);
    async_ld_b128(lds_off(&Vs[buf][16 + lr0][c0]), vbase + (size_t)(s0 + 16 + lr0) * HD_ + c0);
  };

  const int NITER = S_ / 32;   // 64
  issue_tile(0, 0);
  for (int it = 0; it < NITER; ++it) {
    const int buf = it & 1;
    if (it + 1 < NITER) { issue_tile((it + 1) * 32, buf ^ 1); wait_async_le(4); }
    else                { wait_async_le(0); }
    __syncthreads();

    // scores: S = Q * K^T over hd=64 (16 WMMAs per 16x16 tile, 2 tiles)
    v8f st0 = {}, st1 = {};
#pragma unroll
    for (int kk = 0; kk < 16; kk++) {
      const int kd = kk * 4 + kb;
      v2f a;  a.x  = qa[kk].x;                 a.y  = qa[kk].y;
      v2f b0; b0.x = Ks[buf][n16][kd];         b0.y = Ks[buf][n16][kd + 1];
      v2f b1; b1.x = Ks[buf][16 + n16][kd];    b1.y = Ks[buf][16 + n16][kd + 1];
      st0 = wmma4(a, b0, st0);
      st1 = wmma4(a, b1, st1);
    }

    // online softmax (row reductions within 16-lane halves match C layout)
    float alpha_r[8];
#pragma unroll
    for (int r = 0; r < 8; r++) {
      const float s0v = st0[r] * 0.125f;     // 1/sqrt(64)
      const float s1v = st1[r] * 0.125f;
      float mx = fmaxf(s0v, s1v);
#pragma unroll
      for (int d = 8; d >= 1; d >>= 1) mx = fmaxf(mx, __shfl_xor(mx, d, 32));
      const float mnew = fmaxf(m_i[r], mx);
      const float p0 = __expf(s0v - mnew);
      const float p1 = __expf(s1v - mnew);
      float rs = p0 + p1;
#pragma unroll
      for (int d = 8; d >= 1; d >>= 1) rs += __shfl_xor(rs, d, 32);
      alpha_r[r] = __expf(m_i[r] - mnew);
      l_i[r] = l_i[r] * alpha_r[r] + rs;
      m_i[r] = mnew;
      Ps[wave][r + 8 * half][n16]      = p0;   // relay P through LDS
      Ps[wave][r + 8 * half][16 + n16] = p1;
    }
#pragma unroll
    for (int ct = 0; ct < 4; ct++)
#pragma unroll
      for (int r = 0; r < 8; r++) acc[ct][r] *= alpha_r[r];

    __syncthreads();

    // acc += P(16x32) * V(32x64): 4 col-tiles x 8 k-steps
#pragma unroll
    for (int ct = 0; ct < 4; ct++) {
#pragma unroll
      for (int kk = 0; kk < 8; kk++) {
        const int k = kk * 4 + kb;
        v2f a;  a.x  = Ps[wave][n16][k];            a.y  = Ps[wave][n16][k + 1];
        v2f bb; bb.x = Vs[buf][k][ct * 16 + n16];   bb.y = Vs[buf][k + 1][ct * 16 + n16];
        acc[ct] = wmma4(a, bb, acc[ct]);
      }
    }
    __syncthreads();
  }

  // normalize + write back into QO (same (rows, head-cols) this block read)
#pragma unroll
  for (int r = 0; r < 8; r++) {
    const float inv = 1.0f / l_i[r];
    const int tok = b * T_ + t0 + wave * 16 + r + 8 * half;
    float* orow = QO + (size_t)tok * D_ + h * HD_;
#pragma unroll
    for (int ct = 0; ct < 4; ct++) orow[ct * 16 + n16] = acc[ct][r] * inv;
  }
}

// ---------------------------------------------------------------------------
// Host-side launcher
// ---------------------------------------------------------------------------
extern "C" void kernel_launch(void* const* d_in, const int* in_sizes, int n_in,
                              void* d_out, int out_size, void* d_ws, size_t ws_size,
                              hipStream_t stream) {
  const float* x       = (const float*)d_in[0];   // [2,1024,1024]
  const float* k_cache = (const float*)d_in[1];   // [2,4,1024,64]
  const float* v_cache = (const float*)d_in[2];   // [2,4,1024,64]
  const int*   pos     = (const int*)  d_in[3];   // [2,1024]
  const float* Wq      = (const float*)d_in[4];   // [1024,1024]
  const float* Wk      = (const float*)d_in[5];   // [1024,256]
  const float* Wv      = (const float*)d_in[6];   // [1024,256]
  const float* Wo      = (const float*)d_in[7];   // [1024,1024]

  float* out   = (float*)d_out;                        // [2,1024,1024]
  float* out_k = out   + (size_t)B_ * T_ * D_;         // [2,4,2048,64]
  float* out_v = out_k + (size_t)B_ * KV_ * S_ * HD_;  // [2,4,2048,64]

  float* Qws = (float*)d_ws;                           // [2048,1024] (also attn out)
  float* Kn  = Qws + (size_t)B_ * T_ * D_;             // [2048,256]
  float* Vn  = Kn  + (size_t)B_ * T_ * (KV_ * HD_);    // [2048,256]

  const int Mtok = B_ * T_;  // 2048

  // 1) projections
  wmma_gemm_f32<<<dim3(D_ / 64, Mtok / 64), 256, 0, stream>>>(x, Wq, Qws, Mtok, D_, D_);
  wmma_gemm_f32<<<dim3((KV_ * HD_) / 64, Mtok / 64), 256, 0, stream>>>(x, Wk, Kn, Mtok, KV_ * HD_, D_);
  wmma_gemm_f32<<<dim3((KV_ * HD_) / 64, Mtok / 64), 256, 0, stream>>>(x, Wv, Vn, Mtok, KV_ * HD_, D_);

  // 2) RoPE + KV assembly
  rope_q_kernel<<<(B_ * T_ * H_ * 32) / 256, 256, 0, stream>>>(Qws, pos);
  rope_kv_kernel<<<(B_ * T_ * KV_ * 32) / 256, 256, 0, stream>>>(Kn, Vn, pos, out_k, out_v);
  cache_copy_kernel<<<(B_ * KV_ * CACHE_ * HD_) / 256, 256, 0, stream>>>(k_cache, v_cache, out_k, out_v);

  // 3) attention (writes O in place of Q)
  attn_fwd_kernel<<<dim3(T_ / 128, H_, B_), 256, 0, stream>>>(Qws, out_k, out_v);

  // 4) output projection
  wmma_gemm_f32<<<dim3(D_ / 64, Mtok / 64), 256, 0, stream>>>(Qws, Wo, out, Mtok, D_, D_);
}